// Downscale_18811956756990
// MI455X (gfx1250) — compile-verified
//
#include <hip/hip_runtime.h>
#include <cstdint>

// Depthwise 4x4 blur + 2x downsample, pad=(1,1):
//   in  : [8,128,512,512] f32   out : [8,128,256,256] f32
// Memory-bound (AI ~1.6 flop/B, roofline ~58us @ 23.3 TB/s) ->
// stage tiles in LDS via CDNA5 async global->LDS data mover, compute the
// 16-tap window on VALU from b64 LDS reads, store float4 results.

#define TILE_OY   32
#define TILE_OX   64
#define IN_ROWS   66              // 2*TILE_OY + 2 input rows (incl. halo)
#define IN_STRIDE 136             // 130 cols used; 136*4B = 544B = 34*16B (b128 aligned)
#define BLOCK     256

__device__ __forceinline__ void async_load_b128(uint32_t lds_byte_addr, const float* gptr) {
    // CDNA5 GLOBAL_LOAD_ASYNC_TO_LDS_B128 (GV mode, tracked by ASYNCcnt)
    asm volatile("global_load_async_to_lds_b128 %0, %1, off"
                 :: "v"(lds_byte_addr), "v"(gptr)
                 : "memory");
}

__device__ __forceinline__ void wait_asynccnt0() {
    asm volatile("s_wait_asynccnt 0x0" ::: "memory");
}

__global__ __launch_bounds__(BLOCK)
void Downscale_18811956756990_kernel(const float* __restrict__ in,
                                     const float* __restrict__ kern,
                                     float* __restrict__ out)
{
    __shared__ float s_in[IN_ROWS * IN_STRIDE];   // 35,904 B -> ~8 WGs/WGP of 320KB

    const int tid   = threadIdx.x;
    const int tile  = blockIdx.x;
    const int tx    = tile & 3;          // 4 tiles of 64 output cols
    const int ty    = (tile >> 2) & 7;   // 8 tiles of 32 output rows
    const int plane = tile >> 5;         // 8*128 = 1024 planes

    const int ox0      = tx * TILE_OX;
    const int oy0      = ty * TILE_OY;
    const int gx0      = ox0 * 2;        // multiple of 128 -> main body always in-bounds in x
    const int gy_start = oy0 * 2 - 1;

    const float* inp = in + (size_t)plane * (512 * 512);

    // ---- main body: 66 rows x 128 cols via async global->LDS (b128/lane) ----
    // One wave iteration == one row (32 lanes x 16B = 512B).  Fixed 9x unroll.
    const uint32_t s_in_base = (uint32_t)(uintptr_t)(&s_in[0]);
    const int lane = tid & 31;
    const int wrow = tid >> 5;           // 0..7 (8 waves)
    const int col  = 4 + lane * 4;       // LDS col 4 <-> global col gx0 (16B aligned)
#pragma unroll
    for (int k = 0; k < 9; ++k) {
        const int r = wrow + k * 8;      // rows 0..65 covered exactly once
        if (r < IN_ROWS) {               // only k==8 is partial
            const int gy = gy_start + r; // row validity (top/bottom zero pad)
            if ((unsigned)gy < 512u) {
                async_load_b128(s_in_base + (uint32_t)((r * IN_STRIDE + col) * 4),
                                inp + (size_t)gy * 512 + gx0 + lane * 4);
            } else {
                float4 z = {0.f, 0.f, 0.f, 0.f};
                *(float4*)(&s_in[r * IN_STRIDE + col]) = z;
            }
        }
    }

    // ---- halo columns: global x = gx0-1 (LDS col 3) and gx0+128 (LDS col 132) ----
    if (tid < IN_ROWS * 2) {
        const int r    = tid >> 1;
        const int side = tid & 1;
        const int gy   = gy_start + r;
        const int gx   = side ? (gx0 + 128) : (gx0 - 1);
        float v = 0.f;
        if ((unsigned)gy < 512u && (unsigned)gx < 512u)
            v = inp[(size_t)gy * 512 + gx];
        s_in[r * IN_STRIDE + (side ? 132 : 3)] = v;
    }

    // 4x4 kernel, pre-flipped (reference uses true convolution); uniform loads
    float kv[4][4];
#pragma unroll
    for (int wi = 0; wi < 4; ++wi)
#pragma unroll
        for (int wj = 0; wj < 4; ++wj)
            kv[wi][wj] = kern[(3 - wi) * 4 + (3 - wj)];

    wait_asynccnt0();      // this wave's async copies landed in LDS
    __syncthreads();       // all waves' copies visible

    // ---- compute: each thread produces 2 quads of 4 consecutive outputs ----
    // Quad q: ly = q>>4, x0 = (q&15)*4.  Outputs x0..x0+3 need LDS cols
    // [3+2*x0 .. 12+2*x0]; read the enclosing 8B-aligned 12-float window
    // starting at even col 2+2*x0 as six ds_load_b64, use indices 1..10.
    float* outp = out + (size_t)plane * (256 * 256);
#pragma unroll
    for (int qq = 0; qq < 2; ++qq) {
        const int q  = tid + qq * BLOCK;     // 0..511
        const int ly = q >> 4;               // 0..31
        const int x0 = (q & 15) << 2;        // 0,4,...,60 (lanes -> consecutive quads)
        float4 acc = {0.f, 0.f, 0.f, 0.f};
#pragma unroll
        for (int wi = 0; wi < 4; ++wi) {
            const float2* row2 =
                (const float2*)(&s_in[(2 * ly + wi) * IN_STRIDE + 2 + 2 * x0]);
            float rv[12];
#pragma unroll
            for (int t = 0; t < 6; ++t) {    // 6x ds_load_b64
                const float2 p = row2[t];
                rv[2 * t]     = p.x;
                rv[2 * t + 1] = p.y;
            }
#pragma unroll
            for (int wj = 0; wj < 4; ++wj) {
                const float w = kv[wi][wj];
                acc.x += w * rv[1 + wj];     // output x0+0: cols 1+wj
                acc.y += w * rv[3 + wj];     // output x0+1: cols 3+wj
                acc.z += w * rv[5 + wj];     // output x0+2: cols 5+wj
                acc.w += w * rv[7 + wj];     // output x0+3: cols 7+wj
            }
        }
        *(float4*)(&outp[(size_t)(oy0 + ly) * 256 + ox0 + x0]) = acc;  // 16B/lane store
    }
}

extern "C" void kernel_launch(void* const* d_in, const int* in_sizes, int n_in,
                              void* d_out, int out_size, void* d_ws, size_t ws_size,
                              hipStream_t stream) {
    const float* data = (const float*)d_in[0];   // [8,128,512,512] f32
    const float* kern = (const float*)d_in[1];   // [4,4] f32
    float* out = (float*)d_out;                  // [8,128,256,256] f32

    const int planes  = 8 * 128;
    const int tiles   = (256 / TILE_OX) * (256 / TILE_OY);  // 4*8 = 32
    dim3 grid(planes * tiles);                               // 32768 blocks
    dim3 block(BLOCK);
    Downscale_18811956756990_kernel<<<grid, block, 0, stream>>>(data, kern, out);
}